// AlphaCompositionShader_86500641341769
// MI455X (gfx1250) — compile-verified
//
#include <hip/hip_runtime.h>

// AlphaCompositionShader for MI455X (gfx1250, wave32).
// Memory-bound: ~360 MB traffic -> ~15.5 us @ 23.3 TB/s. No matrix math.
// Uses CDNA5 GLOBAL_LOAD_ASYNC_TO_LDS_B128 (ASYNCcnt) to stage the 128 B/pixel
// color records through LDS with perfectly coalesced 16 B/lane requests.

#define K_LAYERS  8
#define BKG_DEPTH 100.0f
#define BLOCK     256
#define PITCH_F   36   // floats per pixel in LDS (144 B = 128 B data + 16 B pad -> 2-way bank conflict only)

__global__ __launch_bounds__(BLOCK) void composite_kernel(
    const float* __restrict__ colors,   // (N, 8, 4)
    const float* __restrict__ zbuf,     // (N, 8)
    const int*   __restrict__ labels,   // (N, 8)
    const float* __restrict__ bgc,      // (3,)
    float* __restrict__ out_img,        // (N, 4)
    float* __restrict__ out_depth,      // (N,)
    int*   __restrict__ out_label,      // (N,)
    float* __restrict__ out_human)      // (N, 8, 4)
{
    __shared__ __align__(16) float lds_cols[BLOCK * PITCH_F];

    const int tid  = threadIdx.x;
    const int pix0 = blockIdx.x * BLOCK;   // first pixel of this block (N divisible by BLOCK)
    const int p    = pix0 + tid;

    // ---- Stage this block's color tile (256 pixels * 128 B = 32 KB) into LDS.
    // Chunk c (16 B) of the tile -> pixel c>>3, fragment c&7. Lane t handles
    // chunks t, t+256, ... : each async instruction issues 32 consecutive 16 B
    // chunks per wave = fully coalesced 512 B.
    {
        const unsigned lbase = (unsigned)(size_t)(&lds_cols[0]); // low 32 bits of as(3) addr = LDS byte offset
        const float* gbase = colors + (size_t)pix0 * (K_LAYERS * 4);
#pragma unroll
        for (int j = 0; j < K_LAYERS; ++j) {
            const int c  = tid + j * BLOCK;
            const int pp = c >> 3;
            const int kk = c & 7;
            unsigned laddr = lbase + (unsigned)(pp * (PITCH_F * 4) + kk * 16);
            const float* gaddr = gbase + (size_t)c * 4;
            asm volatile("global_load_async_to_lds_b128 %0, %1, off"
                         :: "v"(laddr), "v"(gaddr) : "memory");
        }
    }
#if __has_builtin(__builtin_amdgcn_s_wait_asynccnt)
    __builtin_amdgcn_s_wait_asynccnt(0);
#else
    asm volatile("s_wait_asynccnt 0x0" ::: "memory");
#endif
    __syncthreads();

    // ---- Per-pixel scalar streams (B128 loads; lines fully consumed).
    float z[K_LAYERS];
    int   lb[K_LAYERS];
    {
        const float4 z0 = *(const float4*)(zbuf + (size_t)p * 8);
        const float4 z1 = *(const float4*)(zbuf + (size_t)p * 8 + 4);
        z[0] = z0.x; z[1] = z0.y; z[2] = z0.z; z[3] = z0.w;
        z[4] = z1.x; z[5] = z1.y; z[6] = z1.z; z[7] = z1.w;
        const int4 l0 = *(const int4*)(labels + (size_t)p * 8);
        const int4 l1 = *(const int4*)(labels + (size_t)p * 8 + 4);
        lb[0] = l0.x; lb[1] = l0.y; lb[2] = l0.z; lb[3] = l0.w;
        lb[4] = l1.x; lb[5] = l1.y; lb[6] = l1.z; lb[7] = l1.w;
    }

    float4 col[K_LAYERS];
#pragma unroll
    for (int k = 0; k < K_LAYERS; ++k)
        col[k] = *(const float4*)(&lds_cols[tid * PITCH_F + k * 4]);

    const float bg0 = bgc[0], bg1 = bgc[1], bg2 = bgc[2];

    // ---- Back-to-front composite scan (k = 7 .. 0).
    float r = bg0, g = bg1, b = bg2, a = 0.0f, d = BKG_DEPTH;
    int lab = -1;
#pragma unroll
    for (int k = K_LAYERS - 1; k >= 0; --k) {
        const float al  = col[k].w;
        const float ial = 1.0f - al;
        r = col[k].x * al + r * ial;
        g = col[k].y * al + g * ial;
        b = col[k].z * al + b * ial;
        a = fmaxf(al, a);
        if (z[k] > 0.0f) d = z[k] * al + d * ial;            // depth only for z > 0
        if (!(z[k] < 0.0f) && (al > 0.5f)) lab = lb[k];      // !(z<0) matches NaN-safe isfinite(labels_f)
    }

    {
        float4 img; img.x = r; img.y = g; img.z = b; img.w = a;
        *(float4*)(out_img + (size_t)p * 4) = img;
        out_depth[p] = d;
        out_label[p] = lab;   // lab==-1 corresponds to reference's lab>K-0.5 -> -1
    }

    // ---- Human images: front-most fragment per label n (smallest k wins).
#pragma unroll
    for (int n = 0; n < K_LAYERS; ++n) {
        float cr = 0.0f, cg = 0.0f, cb = 0.0f, ca = 0.0f;  // default: appended zero fragment
#pragma unroll
        for (int k = K_LAYERS - 1; k >= 0; --k) {
            const bool m = (!(z[k] < 0.0f)) && (lb[k] == n);
            if (m) { cr = col[k].x; cg = col[k].y; cb = col[k].z; ca = col[k].w; }
        }
        const float ica = 1.0f - ca;
        float4 o;
        o.x = cr * ca + bg0 * ica;
        o.y = cg * ca + bg1 * ica;
        o.z = cb * ca + bg2 * ica;
        o.w = ca;
        *(float4*)(out_human + (size_t)p * (K_LAYERS * 4) + n * 4) = o;
    }
}

extern "C" void kernel_launch(void* const* d_in, const int* in_sizes, int n_in,
                              void* d_out, int out_size, void* d_ws, size_t ws_size,
                              hipStream_t stream) {
    const float* colors = (const float*)d_in[0];   // pixel_colors (B,H,W,K,4) f32
    const float* zb     = (const float*)d_in[1];   // zbuf         (B,H,W,K)   f32
    const int*   lbl    = (const int*)  d_in[2];   // pixel_labels (B,H,W,K)   i32
    const float* bgc    = (const float*)d_in[3];   // background_color (3,)    f32

    const int N = in_sizes[1] / K_LAYERS;          // B*H*W = 1,048,576

    float* out       = (float*)d_out;              // outputs concatenated flat:
    float* out_img   = out;                        // (N,4)
    float* out_depth = out + (size_t)4 * N;        // (N,)
    int*   out_label = (int*)(out + (size_t)5 * N);// (N,) int32 bits
    float* out_human = out + (size_t)6 * N;        // (N,8,4)

    composite_kernel<<<dim3(N / BLOCK), dim3(BLOCK), 0, stream>>>(
        colors, zb, lbl, bgc, out_img, out_depth, out_label, out_human);
}